// AirResistance_80960133530263
// MI455X (gfx1250) — compile-verified
//
#include <hip/hip_runtime.h>
#include <hip/hip_bf16.h>

typedef __attribute__((ext_vector_type(2))) float v2f;
typedef __attribute__((ext_vector_type(8))) float v8f;

// out[b, t] = x[b] * r^t as a K=1 outer-product GEMM on V_WMMA_F32_16X16X4_F32.
// Store-bandwidth bound: 512 MB of f32 out @ 23.3 TB/s ~= 22 us floor.
// -> NT stores (output >> 192MB L2), WMMA produces each 16x16 tile in one op.
// STEPS as a template parameter lets the row stride fold into the 24-bit
// global_store immediate offset (v*STEPS*4 < 2^23), killing the per-store
// 64-bit address-add chains seen in the previous round's asm.
template <int STEPS>
__global__ __launch_bounds__(256)
void AirResistance_geom_wmma(const float* __restrict__ x,
                             const float* __restrict__ dt_ptr,
                             float* __restrict__ out,
                             int steps_rt, int nColTiles)
{
    const int steps = (STEPS > 0) ? STEPS : steps_rt;

    const int lane = threadIdx.x & 31;   // wave32
    // Force the wave id into an SGPR so the tile loop is a true scalar loop:
    // EXEC stays all-1s around the WMMA (hardware requirement).
    const int wave = __builtin_amdgcn_readfirstlane(threadIdx.x >> 5);
    const int n    = lane & 15;          // col within tile (also A-row index)
    const int hi   = lane >> 4;          // half-wave select
    const int b0   = blockIdx.x << 4;    // 16 batch rows per block

    // r exactly as the reference forms it (f32); cancellation-free log2:
    // e = r - 1 is exact (Sterbenz); log2(1+e) = e*(1 - e/2 + e^2/3)/ln2.
    const float dt    = dt_ptr[0];
    const float r     = 1.0f + (-0.01f) * dt;          // COEF = 0.99 - 1.0
    const float e     = r - 1.0f;
    const float log2r = e * (1.0f + e * (-0.5f + e * (1.0f / 3.0f)))
                          * 1.44269504088896340736f;

    // A matrix 16x4 f32 (2 VGPRs/lane): only the K=0 slice nonzero = x[b0+m];
    // all other K slots zero -> only K=0 contributes to the MAC sum.
    v2f a;
    const float xv = x[b0 + n];
    a.x = (hi == 0) ? xv : 0.0f;
    a.y = 0.0f;

    // Per-lane store base; per-tile we only add t0, and the 8 row stores use
    // compile-time immediate offsets v*STEPS.
    float* const base = out + (size_t)(b0 + (hi << 3)) * (size_t)steps + (size_t)n;

    for (int ct = wave; ct < nColTiles; ct += 8) {
        const int t0 = ct << 4;

        // B matrix 4x16 f32: only K=0 row nonzero = r^(t0+n).
        const float p = __builtin_exp2f((float)(t0 + n) * log2r);
        v2f b;
        b.x = (hi == 0) ? p : 0.0f;
        b.y = 0.0f;

        v8f c = {};
        // (neg_a, A, neg_b, B, c_mod, C, reuse_a, reuse_b)
        c = __builtin_amdgcn_wmma_f32_16x16x4_f32(false, a, false, b,
                                                  (short)0, c, false, false);

        // D layout: lane, VGPR v -> (M = v + 8*hi, N = n). Row-major out(B,steps).
        float* dst = base + t0;
        #pragma unroll
        for (int v = 0; v < 8; ++v) {
            __builtin_nontemporal_store(c[v], dst + (size_t)v * steps);
        }
    }
}

extern "C" void kernel_launch(void* const* d_in, const int* in_sizes, int n_in,
                              void* d_out, int out_size, void* d_ws, size_t ws_size,
                              hipStream_t stream) {
    // Inputs (reference order): steps (i32 scalar), x (B floats), delta_t (f32 scalar)
    const float* x  = (const float*)d_in[1];
    const float* dt = (const float*)d_in[2];
    float* out      = (float*)d_out;

    const int B         = in_sizes[1];      // 32768
    const int steps     = out_size / B;     // 4096 (derived host-side)
    const int nColTiles = steps >> 4;       // 256

    dim3 grid(B >> 4);                      // 2048 row-blocks of 16
    dim3 block(256);                        // 8 wave32s

    if (steps == 4096) {
        AirResistance_geom_wmma<4096><<<grid, block, 0, stream>>>(
            x, dt, out, steps, nColTiles);
    } else {
        AirResistance_geom_wmma<0><<<grid, block, 0, stream>>>(
            x, dt, out, steps, nColTiles);
    }
}